// PrototypeBasedClassifier_66769561584356
// MI455X (gfx1250) — compile-verified
//
#include <hip/hip_runtime.h>
#include <hip/hip_bf16.h>
#include <math.h>

// Problem constants (match reference)
#define Bn   128
#define Cn   8
#define Dn   1024
#define Hn   8
#define Sn   14            // 2+3+4+5
#define DHn  (Dn / Hn)     // 128
#define NROW (Bn * Sn)     // 1792

typedef float v2f __attribute__((ext_vector_type(2)));
typedef float v8f __attribute__((ext_vector_type(8)));
typedef uint32_t u32x4 __attribute__((ext_vector_type(4)));
typedef uint32_t u32x8 __attribute__((ext_vector_type(8)));

__device__ __forceinline__ float wave_sum32(float v) {
#pragma unroll
  for (int off = 16; off > 0; off >>= 1) v += __shfl_xor(v, off, 32);
  return v;
}

// ---------------------------------------------------------------- init
__global__ void k_init(float* accum) {
  if (threadIdx.x < 4) accum[threadIdx.x] = 0.f;
}

// ------------------------------------------------- prototype argmin + repr
__global__ void k_argmin(const float* __restrict__ x,
                         const float* __restrict__ p2, const float* __restrict__ p3,
                         const float* __restrict__ p4, const float* __restrict__ p5,
                         float* accum, int* cls_all) {
  __shared__ float xs[Dn];
  __shared__ float dist[40];
  const int b = blockIdx.x;
  for (int d = threadIdx.x; d < Dn; d += blockDim.x) xs[d] = x[(size_t)b * Dn + d];
  const float* pgs[4] = {p2, p3, p4, p5};
  float repr = 0.f;
  for (int g = 0; g < 4; ++g) {
    const int K = g + 2, n = Cn * K;
    __syncthreads();
    if ((int)threadIdx.x < n) {
      const float* p = pgs[g] + (size_t)threadIdx.x * Dn;
      float s = 0.f;
      for (int d = 0; d < Dn; ++d) { float t = xs[d] - p[d]; s += t * t; }
      dist[threadIdx.x] = s;
    }
    __syncthreads();
    if (threadIdx.x == 0) {
      float best = dist[0]; int bi = 0;
      for (int j = 1; j < n; ++j) if (dist[j] < best) { best = dist[j]; bi = j; }
      cls_all[b * 4 + g] = bi / K;
      repr += best;
    }
  }
  if (threadIdx.x == 0) atomicAdd(&accum[0], repr);
}

// ------------------------------------------------- gather proto_space (B,S,D)
__global__ void k_gather(const float* __restrict__ p2, const float* __restrict__ p3,
                         const float* __restrict__ p4, const float* __restrict__ p5,
                         const int* __restrict__ cls_all, float* __restrict__ ps) {
  const int idx = blockIdx.x * blockDim.x + threadIdx.x;
  if (idx >= NROW * Dn) return;
  const int d = idx & (Dn - 1);
  const int bs = idx >> 10;
  const int s = bs % Sn, b = bs / Sn;
  int g, row;
  if (s < 2)      { g = 0; row = s; }
  else if (s < 5) { g = 1; row = s - 2; }
  else if (s < 9) { g = 2; row = s - 5; }
  else            { g = 3; row = s - 9; }
  const float* pg = (g == 0) ? p2 : (g == 1) ? p3 : (g == 2) ? p4 : p5;
  const int K = g + 2;
  const int c = cls_all[b * 4 + g];
  ps[idx] = pg[((size_t)(c * K + row)) * Dn + d];
}

// ------------------------------------------------- batch stats: std loss + mean(s,d)
__global__ void k_stats(const float* __restrict__ ps, float* __restrict__ mean_sd,
                        float* accum) {
  __shared__ float red[256];
  const int sd = blockIdx.x * blockDim.x + threadIdx.x;
  float contrib = 0.f;
  if (sd < Sn * Dn) {
    float s = 0.f, s2 = 0.f;
    for (int b = 0; b < Bn; ++b) {
      float v = ps[(size_t)b * Sn * Dn + sd];
      s += v; s2 += v * v;
    }
    const float m = s / (float)Bn;
    mean_sd[sd] = m;
    const float var = (s2 - (float)Bn * m * m) / (float)(Bn - 1);
    contrib = fmaxf(0.f, 1.f - sqrtf(var + 1e-4f));
  }
  red[threadIdx.x] = contrib; __syncthreads();
  for (int st = 128; st > 0; st >>= 1) {
    if ((int)threadIdx.x < st) red[threadIdx.x] += red[threadIdx.x + st];
    __syncthreads();
  }
  if (threadIdx.x == 0) atomicAdd(&accum[1], red[0]);
}

// ------------------------------------------------- center P = ps - mean (for TDM)
__global__ void k_center(const float* __restrict__ ps, const float* __restrict__ mean_sd,
                         float* __restrict__ pc) {
  const int idx = blockIdx.x * blockDim.x + threadIdx.x;
  if (idx >= NROW * Dn) return;
  const int d = idx & (Dn - 1);
  const int n = idx >> 10;
  pc[idx] = ps[idx] - mean_sd[(n % Sn) * Dn + d];
}

// ------------------------------------------------- covariance GEMM (WMMA + TDM)
// Cov = P^T P / (N-1); panels fetched by the Tensor Data Mover into padded LDS.
__global__ void k_cov(const float* __restrict__ pc, float* accum) {
  __shared__ float Ap[32][33];
  __shared__ float Bp[32][33];
  const int i0 = blockIdx.x * 32, j0 = blockIdx.y * 32;
  const int tid = threadIdx.x;
  const int wave = tid >> 5, lane = tid & 31;
  const int til = (wave >> 1) * 16, tjl = (wave & 1) * 16;
  const int l15 = lane & 15, kh = lane >> 4;

  // D# group1: data_size=4B, pad_enable, pad_interval=32 dwords, pad_amount=1 dword,
  // tensor_dim0=Dn, tensor_dim1=NROW, tile 32x32, dim0_stride=Dn elements.
  u32x8 g1;
  g1[0] = (2u << 16) | (1u << 20) | (4u << 22);        // 0x01120000
  g1[1] = ((uint32_t)Dn & 0xFFFFu) << 16;              // tensor_dim0 lo16
  g1[2] = ((uint32_t)Dn >> 16) | (((uint32_t)NROW & 0xFFFFu) << 16);
  g1[3] = ((uint32_t)NROW >> 16) | (32u << 16);        // tile_dim0 = 32
  g1[4] = 32u;                                         // tile_dim1 = 32
  g1[5] = (uint32_t)Dn;                                // dim0 stride (elements)
  g1[6] = 0u;
  g1[7] = 0u;

  const uint32_t aLds = (uint32_t)(size_t)(&Ap[0][0]);
  const uint32_t bLds = (uint32_t)(size_t)(&Bp[0][0]);

  v8f acc = {};
  for (int n0 = 0; n0 < NROW; n0 += 32) {
    __syncthreads();                       // everyone done with previous panels
    if (wave == 0) {
      const uint64_t gaA = (uint64_t)(size_t)(pc + (size_t)n0 * Dn + i0);
      const uint64_t gaB = (uint64_t)(size_t)(pc + (size_t)n0 * Dn + j0);
      u32x4 g0a, g0b;
      g0a[0] = 1u; g0a[1] = aLds;
      g0a[2] = (uint32_t)gaA;
      g0a[3] = ((uint32_t)(gaA >> 32) & 0x01FFFFFFu) | 0x80000000u;  // type=2
      g0b[0] = 1u; g0b[1] = bLds;
      g0b[2] = (uint32_t)gaB;
      g0b[3] = ((uint32_t)(gaB >> 32) & 0x01FFFFFFu) | 0x80000000u;
      asm volatile("tensor_load_to_lds %0, %1" :: "s"(g0a), "s"(g1) : "memory");
      asm volatile("tensor_load_to_lds %0, %1" :: "s"(g0b), "s"(g1) : "memory");
      __builtin_amdgcn_s_wait_tensorcnt(0);
    }
    __syncthreads();                       // panels visible to all waves
    v2f a[8], bv[8];
#pragma unroll
    for (int s8 = 0; s8 < 8; ++s8) {
      const int kk = s8 * 4;
      a[s8].x  = Ap[kk + 2 * kh][til + l15];
      a[s8].y  = Ap[kk + 2 * kh + 1][til + l15];
      bv[s8].x = Bp[kk + 2 * kh][tjl + l15];
      bv[s8].y = Bp[kk + 2 * kh + 1][tjl + l15];
    }
#pragma unroll
    for (int s8 = 0; s8 < 8; ++s8)
      acc = __builtin_amdgcn_wmma_f32_16x16x4_f32(false, a[s8], false, bv[s8],
                                                  (short)0, acc, false, false);
  }
  const float invN = 1.0f / (float)(NROW - 1);
  float local = 0.f;
#pragma unroll
  for (int r = 0; r < 8; ++r) {
    const int m = r + 8 * kh;
    const int gi = i0 + til + m;
    const int gj = j0 + tjl + l15;
    const float cv = acc[r] * invN;
    if (gi != gj) local += cv * cv;
  }
  local = wave_sum32(local);
  if (lane == 0) atomicAdd(&accum[2], local);
}

// ------------------------------------------------- LayerNorm with routed gamma/beta
__global__ void k_ln(const float* __restrict__ ps, const int* __restrict__ cls_all,
                     const float* __restrict__ ln_g, const float* __restrict__ ln_b,
                     float* __restrict__ hbuf) {
  __shared__ float red[256];
  __shared__ float stats[2];
  const int bs = blockIdx.x;
  const int b = bs / Sn;
  const float* row = ps + (size_t)bs * Dn;
  float s = 0.f, s2 = 0.f;
  for (int d = threadIdx.x; d < Dn; d += 256) { float v = row[d]; s += v; s2 += v * v; }
  red[threadIdx.x] = s; __syncthreads();
  for (int st = 128; st > 0; st >>= 1) {
    if ((int)threadIdx.x < st) red[threadIdx.x] += red[threadIdx.x + st];
    __syncthreads();
  }
  if (threadIdx.x == 0) stats[0] = red[0] / (float)Dn;
  __syncthreads();
  red[threadIdx.x] = s2; __syncthreads();
  for (int st = 128; st > 0; st >>= 1) {
    if ((int)threadIdx.x < st) red[threadIdx.x] += red[threadIdx.x + st];
    __syncthreads();
  }
  if (threadIdx.x == 0) {
    const float mu = stats[0];
    const float var = red[0] / (float)Dn - mu * mu;
    stats[1] = rsqrtf(var + 1e-5f);
  }
  __syncthreads();
  const float mu = stats[0], rs = stats[1];
  const int c = cls_all[b * 4 + 3];
  for (int d = threadIdx.x; d < Dn; d += 256)
    hbuf[(size_t)bs * Dn + d] = (row[d] - mu) * rs * ln_g[c * Dn + d] + ln_b[c * Dn + d];
}

// ------------------------------------------------- qkv GEMM (WMMA): h @ Wqkv^T + b
__global__ void k_qkv(const float* __restrict__ hbuf, const int* __restrict__ cls_all,
                      const float* __restrict__ w_qkv, const float* __restrict__ b_qkv,
                      float* __restrict__ qkv) {
  __shared__ float As[32][17];
  const int b = blockIdx.x;
  const int c = cls_all[b * 4 + 3];
  const int f0 = blockIdx.y * 64;
  const int tid = threadIdx.x, wave = tid >> 5, lane = tid & 31;
  const int fw = f0 + wave * 16;
  const int l15 = lane & 15, kh = lane >> 4;
  const float* hb = hbuf + (size_t)b * Sn * Dn;
  const float2* w2 = (const float2*)(w_qkv + (size_t)c * 3 * Dn * Dn + (size_t)(fw + l15) * Dn);
  v8f acc = {};
  for (int k0 = 0; k0 < Dn; k0 += 32) {
    __syncthreads();
    {
      const int m = tid & 15, kr = tid >> 4;
#pragma unroll
      for (int r = 0; r < 4; ++r) {
        const int k = kr + 8 * r;
        As[k][m] = (m < Sn) ? hb[(size_t)m * Dn + k0 + k] : 0.f;
      }
    }
    __syncthreads();
    v2f a[8], bb[8];
#pragma unroll
    for (int s8 = 0; s8 < 8; ++s8) {
      const int kk = s8 * 4;
      a[s8].x = As[kk + 2 * kh][l15];
      a[s8].y = As[kk + 2 * kh + 1][l15];
      const float2 wv = w2[(k0 + kk) / 2 + kh];
      bb[s8].x = wv.x; bb[s8].y = wv.y;
    }
    if (k0 + 32 < Dn) __builtin_prefetch(w2 + (k0 + 32) / 2, 0, 0);
#pragma unroll
    for (int s8 = 0; s8 < 8; ++s8)
      acc = __builtin_amdgcn_wmma_f32_16x16x4_f32(false, a[s8], false, bb[s8],
                                                  (short)0, acc, false, false);
  }
#pragma unroll
  for (int r = 0; r < 8; ++r) {
    const int m = r + 8 * kh;
    if (m < Sn) {
      const int f = fw + l15;
      qkv[((size_t)b * Sn + m) * 3 * Dn + f] = acc[r] + b_qkv[c * 3 * Dn + f];
    }
  }
}

// ------------------------------------------------- tiny 14x14 softmax attention
__global__ void k_attn(const float* __restrict__ qkv, float* __restrict__ obuf) {
  __shared__ float sc[Sn][Sn];
  const int b = blockIdx.x / Hn, hh = blockIdx.x % Hn;
  const float* qb = qkv + (size_t)b * Sn * 3 * Dn;
  const float scale = rsqrtf((float)DHn);
  const int t = threadIdx.x;
  if (t < Sn * Sn) {
    const int i = t / Sn, j = t % Sn;
    const float* qi = qb + (size_t)i * 3 * Dn + hh * DHn;
    const float* kj = qb + (size_t)j * 3 * Dn + Dn + hh * DHn;
    float s = 0.f;
    for (int e = 0; e < DHn; ++e) s += qi[e] * kj[e];
    sc[i][j] = s * scale;
  }
  __syncthreads();
  if (t < Sn) {
    float mx = sc[t][0];
    for (int j = 1; j < Sn; ++j) mx = fmaxf(mx, sc[t][j]);
    float sum = 0.f;
    for (int j = 0; j < Sn; ++j) { float e = __expf(sc[t][j] - mx); sc[t][j] = e; sum += e; }
    const float inv = 1.f / sum;
    for (int j = 0; j < Sn; ++j) sc[t][j] *= inv;
  }
  __syncthreads();
  for (int idx = t; idx < Sn * DHn; idx += 256) {
    const int i = idx / DHn, e = idx % DHn;
    float s = 0.f;
    for (int j = 0; j < Sn; ++j)
      s += sc[i][j] * qb[(size_t)j * 3 * Dn + 2 * Dn + hh * DHn + e];
    obuf[((size_t)b * Sn + i) * Dn + hh * DHn + e] = s;
  }
}

// ------------------------------------------------- proj GEMM (WMMA) + bias + residual
__global__ void k_proj(const float* __restrict__ obuf, const float* __restrict__ ps,
                       const int* __restrict__ cls_all,
                       const float* __restrict__ w_proj, const float* __restrict__ b_proj,
                       float* __restrict__ att) {
  __shared__ float As[32][17];
  const int b = blockIdx.x;
  const int c = cls_all[b * 4 + 3];
  const int e0 = blockIdx.y * 64;
  const int tid = threadIdx.x, wave = tid >> 5, lane = tid & 31;
  const int ew = e0 + wave * 16;
  const int l15 = lane & 15, kh = lane >> 4;
  const float* ob = obuf + (size_t)b * Sn * Dn;
  const float2* w2 = (const float2*)(w_proj + (size_t)c * Dn * Dn + (size_t)(ew + l15) * Dn);
  v8f acc = {};
  for (int k0 = 0; k0 < Dn; k0 += 32) {
    __syncthreads();
    {
      const int m = tid & 15, kr = tid >> 4;
#pragma unroll
      for (int r = 0; r < 4; ++r) {
        const int k = kr + 8 * r;
        As[k][m] = (m < Sn) ? ob[(size_t)m * Dn + k0 + k] : 0.f;
      }
    }
    __syncthreads();
    v2f a[8], bb[8];
#pragma unroll
    for (int s8 = 0; s8 < 8; ++s8) {
      const int kk = s8 * 4;
      a[s8].x = As[kk + 2 * kh][l15];
      a[s8].y = As[kk + 2 * kh + 1][l15];
      const float2 wv = w2[(k0 + kk) / 2 + kh];
      bb[s8].x = wv.x; bb[s8].y = wv.y;
    }
    if (k0 + 32 < Dn) __builtin_prefetch(w2 + (k0 + 32) / 2, 0, 0);
#pragma unroll
    for (int s8 = 0; s8 < 8; ++s8)
      acc = __builtin_amdgcn_wmma_f32_16x16x4_f32(false, a[s8], false, bb[s8],
                                                  (short)0, acc, false, false);
  }
#pragma unroll
  for (int r = 0; r < 8; ++r) {
    const int m = r + 8 * kh;
    if (m < Sn) {
      const int e = ew + l15;
      const size_t off = ((size_t)b * Sn + m) * Dn + e;
      att[off] = acc[r] + b_proj[c * Dn + e] + ps[off];
    }
  }
}

// ------------------------------------------------- normalize rows, pool, cls head
__global__ void k_pool(const float* __restrict__ att, const float* __restrict__ w_cls,
                       const float* __restrict__ b_cls, float* __restrict__ out) {
  __shared__ float pooled[Dn];
  __shared__ float red[256];
  __shared__ float sinv;
  const int b = blockIdx.x, t = threadIdx.x;
  for (int d = t; d < Dn; d += 256) pooled[d] = 0.f;
  __syncthreads();
  for (int s = 0; s < Sn; ++s) {
    const float* row = att + ((size_t)b * Sn + s) * Dn;
    float ss = 0.f;
    for (int d = t; d < Dn; d += 256) { float v = row[d]; ss += v * v; }
    red[t] = ss; __syncthreads();
    for (int st = 128; st > 0; st >>= 1) {
      if (t < st) red[t] += red[t + st];
      __syncthreads();
    }
    if (t == 0) sinv = 1.f / fmaxf(sqrtf(red[0]), 1e-12f);
    __syncthreads();
    for (int d = t; d < Dn; d += 256) pooled[d] += row[d] * sinv;
    __syncthreads();
  }
  for (int d = t; d < Dn; d += 256) pooled[d] *= (1.f / (float)Sn);
  __syncthreads();
  for (int c = 0; c < Cn; ++c) {
    float s = 0.f;
    for (int d = t; d < Dn; d += 256) s += pooled[d] * w_cls[c * Dn + d];
    red[t] = s; __syncthreads();
    for (int st = 128; st > 0; st >>= 1) {
      if (t < st) red[t] += red[t + st];
      __syncthreads();
    }
    if (t == 0) out[b * Cn + c] = red[0] + b_cls[c];
    __syncthreads();
  }
}

// ------------------------------------------------- finalize scalar vicreg loss
__global__ void k_final(const float* accum, float* out) {
  if (threadIdx.x == 0) {
    const float repr = accum[0] / (float)(Bn * 4 * Dn);
    const float stdl = accum[1] / (float)(Sn * Dn);
    const float covl = accum[2] / (float)Dn;
    out[Bn * Cn] = 25.f * repr + 25.f * stdl + covl;
  }
}

extern "C" void kernel_launch(void* const* d_in, const int* in_sizes, int n_in,
                              void* d_out, int out_size, void* d_ws, size_t ws_size,
                              hipStream_t stream) {
  const float* x    = (const float*)d_in[0];
  const float* p2   = (const float*)d_in[1];
  const float* p3   = (const float*)d_in[2];
  const float* p4   = (const float*)d_in[3];
  const float* p5   = (const float*)d_in[4];
  const float* ln_g = (const float*)d_in[5];
  const float* ln_b = (const float*)d_in[6];
  const float* wqkv = (const float*)d_in[7];
  const float* bqkv = (const float*)d_in[8];
  const float* wprj = (const float*)d_in[9];
  const float* bprj = (const float*)d_in[10];
  const float* wcls = (const float*)d_in[11];
  const float* bcls = (const float*)d_in[12];
  float* out = (float*)d_out;

  // workspace layout (floats)
  float* ws      = (float*)d_ws;
  float* accum   = ws;                          // [0]=repr [1]=std [2]=cov
  int*   cls_all = (int*)(ws + 8);              // B*4 ints
  float* ps      = ws + 1024;                   // proto_space   (B,S,D)
  float* mean    = ps + (size_t)NROW * Dn;      // (S,D)
  float* hbuf    = mean + (size_t)Sn * Dn;      // LN output     (B,S,D)
  float* qkv     = hbuf + (size_t)NROW * Dn;    // (B,S,3D)
  float* obuf    = qkv + (size_t)NROW * 3 * Dn; // attn output   (B,S,D)
  float* pc      = obuf + (size_t)NROW * Dn;    // centered P    (B*S,D)
  float* att     = hbuf;                        // reuse h region for attended

  k_init  <<<1, 32, 0, stream>>>(accum);
  k_argmin<<<Bn, 128, 0, stream>>>(x, p2, p3, p4, p5, accum, cls_all);
  k_gather<<<(NROW * Dn + 255) / 256, 256, 0, stream>>>(p2, p3, p4, p5, cls_all, ps);
  k_stats <<<(Sn * Dn + 255) / 256, 256, 0, stream>>>(ps, mean, accum);
  k_center<<<(NROW * Dn + 255) / 256, 256, 0, stream>>>(ps, mean, pc);
  k_cov   <<<dim3(Dn / 32, Dn / 32), 128, 0, stream>>>(pc, accum);
  k_ln    <<<NROW, 256, 0, stream>>>(ps, cls_all, ln_g, ln_b, hbuf);
  k_qkv   <<<dim3(Bn, 3 * Dn / 64), 128, 0, stream>>>(hbuf, cls_all, wqkv, bqkv, qkv);
  k_attn  <<<Bn * Hn, 256, 0, stream>>>(qkv, obuf);
  k_proj  <<<dim3(Bn, Dn / 64), 128, 0, stream>>>(obuf, ps, cls_all, wprj, bprj, att);
  k_pool  <<<Bn, 256, 0, stream>>>(att, wcls, bcls, out);
  k_final <<<1, 32, 0, stream>>>(accum, out);
}